// VectorQuantizer_38130719654178
// MI455X (gfx1250) — compile-verified
//
#include <hip/hip_runtime.h>
#include <cstdint>
#include <cstddef>

typedef float v2f __attribute__((ext_vector_type(2)));
typedef float v8f __attribute__((ext_vector_type(8)));
typedef int   v4i __attribute__((vector_size(16)));

#define B_ROWS      32768
#define KCODES      4096
#define DDIM        256
#define ROWS_PER_WG 128                      /* 8 waves x 16 rows */
#define WAVE_ROWS   16
#define NWAVES      8
#define NT_TILES    (KCODES / 16)            /* 256 column tiles */
#define NUM_WGS     (B_ROWS / ROWS_PER_WG)   /* 256 */
#define LDS_PITCH   260                      /* 256 + 4 pad: bank-conflict-free */
#define ZQ_ELEMS    ((size_t)B_ROWS * DDIM)  /* 8388608 */
#define IDX_OFF     ZQ_ELEMS
#define LOSS_OFF    (ZQ_ELEMS + B_ROWS)

// ---------------------------------------------------------------------------
// CDNA5 async global->LDS copy (no VGPR staging, tracked by ASYNCcnt).
// Builtin signature (from compiler diagnostic): (v4i addrspace(1)*,
// v4i addrspace(3)*, imm offset, imm cpol).
// ---------------------------------------------------------------------------
typedef __attribute__((address_space(1))) v4i* g_v4i_p;
typedef __attribute__((address_space(3))) v4i* l_v4i_p;

#if __has_builtin(__builtin_amdgcn_global_load_async_to_lds_b128)
#define HAVE_ASYNC_LDS 1
__device__ __forceinline__ void async_copy_b128(const float* gsrc, float* lds) {
    __builtin_amdgcn_global_load_async_to_lds_b128(
        (g_v4i_p)gsrc, (l_v4i_p)lds, 0, 0);
}
#else
#define HAVE_ASYNC_LDS 0
#warning "global_load_async_to_lds_b128 builtin unavailable; using sync copy"
__device__ __forceinline__ void async_copy_b128(const float* gsrc, float* lds) {
    *(float4*)lds = *(const float4*)gsrc;   // fallback: load+ds_store
}
#endif

__device__ __forceinline__ void wait_async_zero() {
#if HAVE_ASYNC_LDS
#if __has_builtin(__builtin_amdgcn_s_wait_asynccnt)
    __builtin_amdgcn_s_wait_asynccnt(0);
#else
    asm volatile("s_wait_asynccnt 0x0" ::: "memory");
#endif
#endif
}

// Map float -> monotonically ordered unsigned (handles negatives).
__device__ __forceinline__ unsigned sortable_f32(float f) {
    unsigned u = __float_as_uint(f);
    return u ^ (((int)u >> 31) | 0x80000000u);
}

// ---------------------------------------------------------------------------
// Kernel 1: c_sq[k] = sum_d codebook[k][d]^2
// ---------------------------------------------------------------------------
__global__ void vq_csq_kernel(const float* __restrict__ cb,
                              float* __restrict__ csq) {
    int k = blockIdx.x * blockDim.x + threadIdx.x;
    if (k >= KCODES) return;
    const float4* row = (const float4*)(cb + (size_t)k * DDIM);
    float s = 0.f;
#pragma unroll 8
    for (int i = 0; i < DDIM / 4; ++i) {
        float4 v = row[i];
        s += v.x * v.x + v.y * v.y + v.z * v.z + v.w * v.w;
    }
    csq[k] = s;
}

// ---------------------------------------------------------------------------
// Kernel 2: 128 rows per WG; each wave owns 16 rows (A in 128 VGPRs).
// B tiles (16 codes x 256) double-buffered in LDS via async global->LDS
// copies issued one tile ahead; WMMAs only ever wait on short LDS latency.
// ---------------------------------------------------------------------------
__global__ __launch_bounds__(256) void vq_main_kernel(
    const float* __restrict__ ze, const float* __restrict__ cb,
    const float* __restrict__ csq, float* __restrict__ out,
    float* __restrict__ partials) {
    __shared__ float btile[2][WAVE_ROWS][LDS_PITCH];
    __shared__ int rowidx[ROWS_PER_WG];
    __shared__ float wsum[NWAVES];

    const int tid  = threadIdx.x;
    const int wave = tid >> 5;
    const int lane = tid & 31;
    const int lo   = lane & 15;   // M (A rows) / N (B,C cols) within tile
    const int hi   = lane >> 4;   // K pair {0,1} vs {2,3}
    const int row0 = blockIdx.x * ROWS_PER_WG;
    const int wrow0 = row0 + wave * WAVE_ROWS;

    // This thread's share of the cooperative B-tile copy (4 x 16B groups).
    const int rr0 = tid >> 6;           // tile rows rr0, rr0+4, rr0+8, rr0+12
    const int c4  = tid & 63;           // float4 column

    // ---- A tile: this wave's 16 rows x 256 dims of z_e, in registers ----
    const float* zrow = ze + (size_t)(wrow0 + lo) * DDIM + 2 * hi;
    v2f a[64];
#pragma unroll
    for (int j = 0; j < 64; ++j)
        a[j] = *(const v2f*)(zrow + 4 * j);

    // ---- prologue: async-copy tile 0 into buffer 0 ----
#pragma unroll
    for (int q = 0; q < 4; ++q) {
        int rr = rr0 + 4 * q;
        async_copy_b128(cb + (size_t)rr * DDIM + 4 * c4,
                        &btile[0][rr][4 * c4]);
    }
    wait_async_zero();
    __syncthreads();

    unsigned long long best[8];
#pragma unroll
    for (int r = 0; r < 8; ++r) best[r] = ~0ull;

    for (int i = 0; i < NT_TILES; ++i) {
        const int cur = i & 1;

        // Issue next tile's async copies; they land in LDS during compute.
        if (i + 1 < NT_TILES) {
            const float* nb = cb + (size_t)(i + 1) * 16 * DDIM;
#pragma unroll
            for (int q = 0; q < 4; ++q) {
                int rr = rr0 + 4 * q;
                async_copy_b128(nb + (size_t)rr * DDIM + 4 * c4,
                                &btile[cur ^ 1][rr][4 * c4]);
            }
        }

        // ---- compute: 64 ds_load_b64 + 64 WMMA against shared B tile ----
        const int col0 = i * 16;
        const float cs = csq[col0 + lo];
        const float* bb = &btile[cur][lo][2 * hi];

        v8f acc0 = {0.f, 0.f, 0.f, 0.f, 0.f, 0.f, 0.f, 0.f};
        v8f acc1 = {0.f, 0.f, 0.f, 0.f, 0.f, 0.f, 0.f, 0.f};
#pragma unroll
        for (int j = 0; j < 64; j += 2) {
            v2f b0 = *(const v2f*)(bb + 4 * j);
            v2f b1 = *(const v2f*)(bb + 4 * j + 4);
            acc0 = __builtin_amdgcn_wmma_f32_16x16x4_f32(
                false, a[j], false, b0, (short)0, acc0, false, false);
            acc1 = __builtin_amdgcn_wmma_f32_16x16x4_f32(
                false, a[j + 1], false, b1, (short)0, acc1, false, false);
        }

        const unsigned idx = (unsigned)(col0 + lo);
#pragma unroll
        for (int r = 0; r < 8; ++r) {
            // per-row-constant z_sq dropped: does not change argmin
            float d = cs - 2.0f * (acc0[r] + acc1[r]);
            unsigned long long key =
                ((unsigned long long)sortable_f32(d) << 32) | idx;
            if (key < best[r]) best[r] = key;
        }

        // Ensure my async copies landed; barrier makes the tile visible to
        // (and orders buffer reuse across) all waves.
        if (i + 1 < NT_TILES) wait_async_zero();
        __syncthreads();
    }

    // ---- wave-local argmin finalize (each wave owns its 16 rows) ----
#pragma unroll
    for (int r = 0; r < 8; ++r) {
        unsigned long long k = best[r];
#pragma unroll
        for (int m = 1; m < 16; m <<= 1) {
            unsigned long long o = __shfl_xor(k, m, 32);
            if (o < k) k = o;
        }
        best[r] = k;
    }
    if (lane == 0 || lane == 16) {
        const int base = wave * WAVE_ROWS + hi * 8;
#pragma unroll
        for (int r = 0; r < 8; ++r) {
            int id = (int)(best[r] & 0xffffffffu);
            rowidx[base + r] = id;
            out[IDX_OFF + row0 + base + r] = (float)id;  // indices output
        }
    }
    __syncthreads();

    // ---- gather z_q (forward value of z_q_st), accumulate loss ----
    float lsum = 0.f;
    for (int e = tid; e < ROWS_PER_WG * DDIM; e += 256) {
        int r = e >> 8;      // row within block
        int c = e & 255;     // column
        float q = cb[(size_t)rowidx[r] * DDIM + c];
        float z = ze[(size_t)(row0 + r) * DDIM + c];
        out[(size_t)(row0 + r) * DDIM + c] = q;
        float df = q - z;
        lsum += df * df;
    }
#pragma unroll
    for (int m = 16; m >= 1; m >>= 1)
        lsum += __shfl_xor(lsum, m, 32);
    if (lane == 0) wsum[wave] = lsum;
    __syncthreads();
    if (tid == 0) {
        float t = 0.f;
#pragma unroll
        for (int w = 0; w < NWAVES; ++w) t += wsum[w];
        partials[blockIdx.x] = t;  // plain store: deterministic
    }
}

// ---------------------------------------------------------------------------
// Kernel 3: vq_loss = 1.25 * sum(partials) / (B*D)   (fixed reduction order)
// ---------------------------------------------------------------------------
__global__ void vq_loss_kernel(const float* __restrict__ partials,
                               float* __restrict__ out) {
    int lane = threadIdx.x;
    float s = 0.f;
    for (int i = lane; i < NUM_WGS; i += 32) s += partials[i];
#pragma unroll
    for (int m = 16; m >= 1; m >>= 1) s += __shfl_xor(s, m, 32);
    if (lane == 0)
        out[LOSS_OFF] = 1.25f * s / (float)((size_t)B_ROWS * DDIM);
}

extern "C" void kernel_launch(void* const* d_in, const int* in_sizes, int n_in,
                              void* d_out, int out_size, void* d_ws,
                              size_t ws_size, hipStream_t stream) {
    const float* ze = (const float*)d_in[0];  // (32768, 256)
    const float* cb = (const float*)d_in[1];  // (4096, 256)
    float* out = (float*)d_out;
    float* csq = (float*)d_ws;            // 4096 floats
    float* partials = csq + KCODES;       // 256 floats

    vq_csq_kernel<<<KCODES / 256, 256, 0, stream>>>(cb, csq);
    vq_main_kernel<<<NUM_WGS, 256, 0, stream>>>(ze, cb, csq, out, partials);
    vq_loss_kernel<<<1, 32, 0, stream>>>(partials, out);
}